// AdaptivePiecewiseLinear_3564822856233
// MI455X (gfx1250) — compile-verified
//
#include <hip/hip_runtime.h>
#include <hip/hip_bf16.h>

typedef __attribute__((ext_vector_type(2))) float v2f;
typedef __attribute__((ext_vector_type(4))) float v4f;
typedef __attribute__((ext_vector_type(8))) float v8f;

#define NUM_INPUTS  128
#define NUM_OUTPUTS 128
#define NUM_POINTS  16
#define BATCH       256
#define CHUNK_I     32     // i-values staged per phase (512 (b,i) pairs)
#define ROW_STRIDE  20     // padded LDS row stride in floats: 80 B rows -> 16B-aligned
                           // b128 stores, and 20*lane mod 64 -> 16 distinct banks

// out[b,o] = sum_i piecewise_linear(x[b,i]; positions[i,0,:], values[i,o,:])
// Recast as GEMM out = W(256 x 2048) * V(2048 x 128) with hat-basis weights W,
// evaluated with V_WMMA_F32_16X16X4_F32 (fp32 A/B/accumulate).
//
// grid.x = 16 row-bands of 16 batches; 8 waves/block, wave w owns output tile
// [w*16, w*16+16). Weights are built cooperatively once per block per K-chunk
// and shared through LDS in A-fragment layout, so the steady-state loop is
// ds_load_b64 (A) + global_load_b64 (B) + v_wmma only.
__global__ __launch_bounds__(256)
void AdaptivePiecewiseLinear_kernel(const float* __restrict__ x,
                                    const float* __restrict__ positions,
                                    const float* __restrict__ values,
                                    float* __restrict__ out) {
    __shared__ float sW[CHUNK_I * 16 * ROW_STRIDE];   // 40 KB

    const int tid = threadIdx.x;
    const int b0  = blockIdx.x * 16;

    const int  lane = tid & 31;
    const int  wave = tid >> 5;
    const int  o0   = wave * 16;
    const int  bl   = lane & 15;      // A row (M) for this lane; also C/D column (N)
    const bool hi   = lane >= 16;     // second half-wave: A holds K+2, D holds rows 8..15
    const int  koff = hi ? 2 : 0;

    v8f c = {};

    // B-fragment base: values[i][o0+bl][koff + 4*step + {0,1}]
    const float* vwave = values + (size_t)(o0 + bl) * NUM_POINTS + koff;

    for (int ch = 0; ch < NUM_INPUTS / CHUNK_I; ++ch) {
        // ---- phase 1: cooperative hat-weight build into LDS (2 pairs/thread) ----
        for (int q = tid; q < CHUNK_I * 16; q += 256) {
            const int b  = q & 15;        // batch-local
            const int il = q >> 4;        // i within chunk
            const int i  = ch * CHUNK_I + il;

            const float  xv = x[(size_t)(b0 + b) * NUM_INPUTS + i];
            const float* pg = positions + (size_t)i * (NUM_OUTPUTS * NUM_POINTS); // o==0 slice

            alignas(16) float p[NUM_POINTS];
            *(v4f*)&p[0]  = *(const v4f*)(pg + 0);
            *(v4f*)&p[4]  = *(const v4f*)(pg + 4);
            *(v4f*)&p[8]  = *(const v4f*)(pg + 8);
            *(v4f*)&p[12] = *(const v4f*)(pg + 12);

            alignas(16) float w[NUM_POINTS];
#pragma unroll
            for (int k = 0; k < NUM_POINTS; ++k) w[k] = 0.f;
            w[0]              = (xv <  p[0])              ? 1.f : 0.f;  // left clamp
            w[NUM_POINTS - 1] = (xv >= p[NUM_POINTS - 1]) ? 1.f : 0.f;  // right clamp
#pragma unroll
            for (int s = 0; s < NUM_POINTS - 1; ++s) {
                const float t = (xv - p[s]) * __builtin_amdgcn_rcpf(p[s + 1] - p[s]);
                const float m = (xv >= p[s] && xv < p[s + 1]) ? 1.f : 0.f;
                w[s]     += m * (1.f - t);
                w[s + 1] += m * t;
            }

            float* row = &sW[(il * 16 + b) * ROW_STRIDE];
            *(v4f*)(row + 0)  = *(const v4f*)&w[0];
            *(v4f*)(row + 4)  = *(const v4f*)&w[4];
            *(v4f*)(row + 8)  = *(const v4f*)&w[8];
            *(v4f*)(row + 12) = *(const v4f*)&w[12];
        }
        __syncthreads();

        // ---- phase 2: WMMA GEMM over this K-chunk ----
        for (int il = 0; il < CHUNK_I; ++il) {
            const int i = ch * CHUNK_I + il;

            // A fragments from LDS (lanes 0-15: K={0,1}, lanes 16-31: K={2,3} per step)
            const float* arow = &sW[(il * 16 + bl) * ROW_STRIDE + koff];
            const v2f a0 = *(const v2f*)(arow + 0);
            const v2f a1 = *(const v2f*)(arow + 4);
            const v2f a2 = *(const v2f*)(arow + 8);
            const v2f a3 = *(const v2f*)(arow + 12);

            // B fragments straight from global (values is L2-resident)
            const float* vb = vwave + (size_t)i * (NUM_OUTPUTS * NUM_POINTS);
            const v2f bf0 = *(const v2f*)(vb + 0);
            const v2f bf1 = *(const v2f*)(vb + 4);
            const v2f bf2 = *(const v2f*)(vb + 8);
            const v2f bf3 = *(const v2f*)(vb + 12);
            if (i + 1 < NUM_INPUTS)
                __builtin_prefetch(vb + NUM_OUTPUTS * NUM_POINTS, 0, 1);

            c = __builtin_amdgcn_wmma_f32_16x16x4_f32(false, a0, false, bf0, (short)0, c, false, false);
            c = __builtin_amdgcn_wmma_f32_16x16x4_f32(false, a1, false, bf1, (short)0, c, false, false);
            c = __builtin_amdgcn_wmma_f32_16x16x4_f32(false, a2, false, bf2, (short)0, c, false, false);
            c = __builtin_amdgcn_wmma_f32_16x16x4_f32(false, a3, false, bf3, (short)0, c, false, false);
        }
        __syncthreads();
    }

    // 16x16 f32 D layout: VGPR j -> (M = j + (hi?8:0), N = bl)
    const int row0 = b0 + (hi ? 8 : 0);
#pragma unroll
    for (int j = 0; j < 8; ++j) {
        out[(size_t)(row0 + j) * NUM_OUTPUTS + o0 + bl] = c[j];
    }
}

extern "C" void kernel_launch(void* const* d_in, const int* in_sizes, int n_in,
                              void* d_out, int out_size, void* d_ws, size_t ws_size,
                              hipStream_t stream) {
    const float* x         = (const float*)d_in[0];  // (256, 128)
    const float* positions = (const float*)d_in[1];  // (128, 128, 16)
    const float* values    = (const float*)d_in[2];  // (128, 128, 16)
    float* out             = (float*)d_out;          // (256, 128)

    dim3 grid(BATCH / 16);   // 16 row-bands of 16 batches
    dim3 block(256);         // 8 waves -> 8 column tiles of 16 outputs
    AdaptivePiecewiseLinear_kernel<<<grid, block, 0, stream>>>(x, positions, values, out);
}